// FeatureRender_46583215292761
// MI455X (gfx1250) — compile-verified
//
#include <hip/hip_runtime.h>
#include <stdint.h>

// Problem constants (from the reference)
#define BS 4
#define C_ 16
#define H_ 512
#define W_ 768
#define HF 128          // H_/4
#define WF 128          // W_/6
#define DH 256
#define DW 256

// Workspace layout (float* view):
//   wsf[0..63]                : base[b*16+c]  = sum over 24 parts of T_b[b,q,c][0,127]
//   ((int2*)(wsf+64))[flat]   : { corr_bits, feature_base_offset } for flat=(b*24+p)*16+c, 1536 entries
// Total: 64*4 + 1536*8 = 12544 bytes.

__global__ __launch_bounds__(256) void fr_precompute(const float* __restrict__ feature,
                                                     float* __restrict__ wsf) {
    __shared__ float s_corr[1536];
    int2* pairs = (int2*)(wsf + 64);
    const int t = threadIdx.x;
    for (int flat = t; flat < 1536; flat += 256) {
        // flat = (b2*24 + p2)*16 + c2 reinterpreted through (c, bs, 24) order:
        int c_orig = flat / 96;
        int rem    = flat - c_orig * 96;
        int b_orig = rem / 24;
        int p_orig = rem - b_orig * 24;
        int r  = p_orig / 6;          // tile row in the 4x6 atlas
        int s6 = p_orig - r * 6;      // tile col
        int base_off = ((b_orig * C_ + c_orig) * H_ + r * HF) * W_ + s6 * WF;
        float corr = feature[base_off + (WF - 1)];   // tile[0, 127]
        s_corr[flat] = corr;
        pairs[flat] = make_int2(__float_as_int(corr), base_off);
    }
    __syncthreads();
    if (t < 64) {
        const int b = t >> 4, c = t & 15;
        float acc = 0.0f;
        #pragma unroll
        for (int p = 0; p < 24; ++p) acc += s_corr[(b * 24 + p) * C_ + c];
        wsf[t] = acc;  // base[b,c]
    }
}

__global__ __launch_bounds__(256) void fr_render(const float* __restrict__ feature,
                                                 const int* __restrict__ dense_pose,
                                                 const float* __restrict__ wsf,
                                                 float* __restrict__ out) {
    // LDS staging: this block's 256 dense_pose pixels (3 KB), the LUT slice for its
    // (uniform) batch (384 pairs = 3 KB), and the 16 base sums (64 B), all moved with
    // CDNA5 async global->LDS b128 copies under one ASYNCcnt fence.
    __shared__ int   s_dp[768];
    __shared__ int2  s_pairs[384];
    __shared__ float s_base[16];

    const int tid = threadIdx.x;
    const int blk = blockIdx.x;
    const int b   = blk >> 8;                       // 256 blocks per batch (uniform)

    const int*   g_dp  = dense_pose + blk * 768;
    const float* g_lut = wsf + 64 + b * 768;        // 384 int2 = 768 floats for this batch
    const float* g_bs  = wsf + b * 16;

#if defined(__gfx1250__)
    {
        unsigned l_dp  = (unsigned)(size_t)(&s_dp[0]);
        unsigned l_lut = (unsigned)(size_t)(&s_pairs[0]);
        unsigned l_bs  = (unsigned)(size_t)(&s_base[0]);
        if (tid < 192) {                            // 192 * 16B = 3 KB each
            const int4* p0 = (const int4*)g_dp + tid;
            const int4* p1 = (const int4*)g_lut + tid;
            unsigned o = (unsigned)(tid * 16);
            asm volatile("global_load_async_to_lds_b128 %0, %1, off"
                         :: "v"(l_dp + o), "v"(p0) : "memory");
            asm volatile("global_load_async_to_lds_b128 %0, %1, off"
                         :: "v"(l_lut + o), "v"(p1) : "memory");
        }
        if (tid < 4) {                              // 4 * 16B = 64 B base sums
            const int4* p2 = (const int4*)g_bs + tid;
            asm volatile("global_load_async_to_lds_b128 %0, %1, off"
                         :: "v"(l_bs + (unsigned)(tid * 16)), "v"(p2) : "memory");
        }
        asm volatile("s_wait_asynccnt 0x0" ::: "memory");
    }
#else
    for (int k = tid; k < 768; k += 256) {
        s_dp[k] = g_dp[k];
        ((int*)s_pairs)[k] = ((const int*)g_lut)[k];
    }
    if (tid < 16) s_base[tid] = g_bs[tid];
#endif
    __syncthreads();

    const int g = blk * 256 + tid;       // flat pixel id over (b, i, j)
    const int i = (g >> 8) & 255;
    const int j = g & 255;

    const int part = s_dp[tid * 3 + 0];
    const int U    = s_dp[tid * 3 + 1];
    const int V    = s_dp[tid * 3 + 2];

    float* const op = out + (b * (C_ * DH * DW) + i * DW + j);

    if ((part > 0) && (V != 0)) {
        // Active pixel: one data-dependent gather per channel.
        const int u = (U * (HF - 1)) / 255;            // == trunc(U*127/255.0f)
        const int v = ((255 - V) * (WF - 1)) / 255;    // == trunc((255-V)*127/255.0f)
        const int uvoff = u * W_ + v;
        // 16 pairs for (b, part-1); 128B-aligned -> read as 8x int4 (2 channels each)
        const int4* lp4 = (const int4*)(&s_pairs[(part - 1) * 16]);
        #pragma unroll
        for (int c2 = 0; c2 < 8; ++c2) {
            int4 q = lp4[c2];
            float corr0 = __int_as_float(q.x);
            float corr1 = __int_as_float(q.z);
            float t0 = feature[q.y + uvoff];
            float t1 = feature[q.w + uvoff];
            // Output is write-once: non-temporal stores keep L2 free for the
            // L2-resident feature tensor that the gathers depend on.
            __builtin_nontemporal_store(s_base[2 * c2 + 0] - corr0 + t0,
                                        op + (2 * c2 + 0) * (DH * DW));
            __builtin_nontemporal_store(s_base[2 * c2 + 1] - corr1 + t1,
                                        op + (2 * c2 + 1) * (DH * DW));
        }
    } else {
        // Inactive: every part gathers the constant (0,127) texel -> precomputed base sum.
        #pragma unroll
        for (int c = 0; c < C_; ++c) {
            __builtin_nontemporal_store(s_base[c], op + c * (DH * DW));
        }
    }
}

extern "C" void kernel_launch(void* const* d_in, const int* in_sizes, int n_in,
                              void* d_out, int out_size, void* d_ws, size_t ws_size,
                              hipStream_t stream) {
    const float* feature    = (const float*)d_in[0];   // (4,16,512,768) f32
    const int*   dense_pose = (const int*)d_in[1];     // (4,256,256,3) i32
    float* out = (float*)d_out;                        // (4,16,256,256) f32
    float* wsf = (float*)d_ws;                         // >= 12544 bytes used

    fr_precompute<<<1, 256, 0, stream>>>(feature, wsf);
    fr_render<<<(BS * DH * DW) / 256, 256, 0, stream>>>(feature, dense_pose, wsf, out);
}